// DifferentiableGaussianRenderer_34875134444363
// MI455X (gfx1250) — compile-verified
//
#include <hip/hip_runtime.h>
#include <hip/hip_bf16.h>

// ---------------------------------------------------------------------------
// Differentiable Gaussian renderer, MI455X (gfx1250, wave32).
// Strategy: per (pixel,gaussian) pair we need two 3-dots (A = <mu*sinv, d>,
// B = <sinv, d^2>).  Batched as f32 WMMA 16x16x4 GEMM:
//   Arows (M=16) = 8 gaussians x {A-row, B-row},  K=8 (padded),
//   Bcols (N=16) = 16 pixels' feature vectors [dx,dy,dz,dx2,dy2,dz2,0,0].
// D layout puts pixel n's gaussians 0-3 in lane n and 4-7 in lane n+16;
// alpha compositing is associative ((I,T) monoid), so the halves composite
// their own segments and merge via ds_swizzle SWAPX16 per 8-gaussian block.
// ---------------------------------------------------------------------------

typedef float v2f __attribute__((ext_vector_type(2)));
typedef float v8f __attribute__((ext_vector_type(8)));

#define GH 512
#define GW 512
#define NG 256
#define NBLK 32   // 256 gaussians / 8 per WMMA block

__device__ __forceinline__ float fast_rcp(float x) {
#if __has_builtin(__builtin_amdgcn_rcpf)
  return __builtin_amdgcn_rcpf(x);
#else
  return 1.0f / x;
#endif
}
__device__ __forceinline__ float fast_rsq(float x) {
#if __has_builtin(__builtin_amdgcn_rsqf)
  return __builtin_amdgcn_rsqf(x);
#else
  return rsqrtf(x);
#endif
}
__device__ __forceinline__ float fast_exp2(float x) {
#if __has_builtin(__builtin_amdgcn_exp2f)
  return __builtin_amdgcn_exp2f(x);
#else
  return exp2f(x);
#endif
}
__device__ __forceinline__ float swapx16(float x) {
  // ds_swizzle_b32, group-of-32 mode: and=0x1f, or=0, xor=0x10 -> 0x401f
  return __int_as_float(__builtin_amdgcn_ds_swizzle(__float_as_int(x), 0x401f));
}

// ---------------------------------------------------------------------------
// Prep: stable z-sort (O(N^2) rank, N=256 -> trivial), precompute
//   w = mu*sinv, sinv, a0' = sigmoid(op)*exp(-0.5*<sinv,mu^2>), sigmoid colors,
// and emit the A-operand table already in WMMA 16x16x4 f32 A-layout:
//   lane m (<16): VGPR holds K0/K1 (wmma0) and K4/K5 (wmma1)
//   lane m+16   : K2/K3 (wmma0) and K6/K7 (wmma1)
// Table: aop[blk*32 + lane] = {a0.v0, a0.v1, a1.v0, a1.v1}
// ---------------------------------------------------------------------------
__global__ void gsplat_prep_kernel(const float* __restrict__ pos,
                                   const float* __restrict__ lsc,
                                   const float* __restrict__ rop,
                                   const float* __restrict__ col,
                                   float4* __restrict__ aop,   // [NBLK*32]
                                   float4* __restrict__ scal)  // [NG]
{
  const int i = threadIdx.x;
  if (i >= NG) return;
  const float mx = pos[i * 3 + 0], my = pos[i * 3 + 1], mz = pos[i * 3 + 2];
  // scales = exp(ls); sinv = 1/scales^2 = exp(-2*ls)
  const float sx = expf(-2.0f * lsc[i * 3 + 0]);
  const float sy = expf(-2.0f * lsc[i * 3 + 1]);
  const float sz = expf(-2.0f * lsc[i * 3 + 2]);
  const float a0 = 1.0f / (1.0f + expf(-rop[i]));
  const float cr = 1.0f / (1.0f + expf(-col[i * 3 + 0]));
  const float cg = 1.0f / (1.0f + expf(-col[i * 3 + 1]));
  const float cb = 1.0f / (1.0f + expf(-col[i * 3 + 2]));

  // stable rank by z (matches jnp.argsort stable tie-break by index)
  int rank = 0;
  for (int j = 0; j < NG; ++j) {
    const float zj = pos[j * 3 + 2];
    rank += (zj < mz) || (zj == mz && j < i);
  }

  const float wx = mx * sx, wy = my * sy, wz = mz * sz;
  const float C = sx * mx * mx + sy * my * my + sz * mz * mz;
  const float a0p = a0 * expf(-0.5f * C);

  const int b = rank >> 3;      // block
  const int j8 = rank & 7;      // gaussian within block; rows 2*j8 (A), 2*j8+1 (B)
  float4* base = aop + b * 32;
  // A-row (M=2j): [wx,wy,wz,0, 0,0,0,0]
  base[2 * j8 + 0]      = make_float4(wx, wy, 0.0f, 0.0f);  // lane 2j  : K0,K1 | K4,K5
  base[16 + 2 * j8 + 0] = make_float4(wz, 0.0f, 0.0f, 0.0f);// lane 2j+16: K2,K3 | K6,K7
  // B-row (M=2j+1): [0,0,0,sx, sy,sz,0,0]
  base[2 * j8 + 1]      = make_float4(0.0f, 0.0f, sy, sz);
  base[16 + 2 * j8 + 1] = make_float4(0.0f, sx, 0.0f, 0.0f);

  scal[rank] = make_float4(a0p, cr, cg, cb);
}

// ---------------------------------------------------------------------------
// Render: 256 threads = 8 waves, each wave owns 16 consecutive pixels of a row.
// ---------------------------------------------------------------------------
__global__ __launch_bounds__(256) void gsplat_render_kernel(
    const float4* __restrict__ aopG, const float4* __restrict__ scalG,
    float* __restrict__ out)
{
  __shared__ float4 shA[NBLK * 32];  // 16 KB: A-operands, WMMA-ready
  __shared__ float4 shS[NG];         //  4 KB: {a0', r, g, b} sorted

  for (int t = threadIdx.x; t < NBLK * 32; t += 256) shA[t] = aopG[t];
  shS[threadIdx.x] = scalG[threadIdx.x];
  __syncthreads();

  const int lane = threadIdx.x & 31;
  const int wv   = threadIdx.x >> 5;
  const int p    = lane & 15;
  const bool hi  = lane >= 16;
  const int x = blockIdx.x * 128 + wv * 16 + p;
  const int y = blockIdx.y;

  // normalized ray direction for pixel (x,y)
  const float ux = ((float)x + 0.5f - 256.0f) * (1.0f / 500.0f);
  const float uy = ((float)y + 0.5f - 256.0f) * (1.0f / 500.0f);
  const float rn = fast_rsq(ux * ux + uy * uy + 1.0f);
  const float dx = ux * rn, dy = uy * rn, dz = rn;

  // B-matrix operands (K x N feature matrix), 16x16x4 B-layout:
  //   VGPR0: K0 (lanes 0-15) | K2 (lanes 16-31);  VGPR1: K1 | K3
  v2f b0, b1;
  b0.x = hi ? dz       : dx;       // K0=dx   | K2=dz
  b0.y = hi ? dx * dx  : dy;       // K1=dy   | K3=dx^2
  b1.x = hi ? 0.0f     : dy * dy;  // K4=dy^2 | K6=0
  b1.y = hi ? 0.0f     : dz * dz;  // K5=dz^2 | K7=0

  float runR = 0.0f, runG = 0.0f, runB = 0.0f, runT = 1.0f;

  for (int blk = 0; blk < NBLK; ++blk) {
    const float4 at = shA[blk * 32 + lane];       // ds_load_b128
    v2f a0v; a0v.x = at.x; a0v.y = at.y;
    v2f a1v; a1v.x = at.z; a1v.y = at.w;

    v8f acc = {0, 0, 0, 0, 0, 0, 0, 0};
    acc = __builtin_amdgcn_wmma_f32_16x16x4_f32(false, a0v, false, b0,
                                                (short)0, acc, false, false);
    acc = __builtin_amdgcn_wmma_f32_16x16x4_f32(false, a1v, false, b1,
                                                (short)0, acc, false, false);
    // lane (p, half h): acc[2i] = A, acc[2i+1] = B for gaussian blk*8 + 4h + i

    float sIr = 0.0f, sIg = 0.0f, sIb = 0.0f, sT = 1.0f;
    const int gbase = blk * 8 + (hi ? 4 : 0);
#pragma unroll
    for (int i = 0; i < 4; ++i) {
      const float Av = acc[2 * i];
      const float Bv = acc[2 * i + 1];
      const float4 sc = shS[gbase + i];           // ds_load_b128
      const float rB  = fast_rcp(Bv + 1e-10f);
      const float tau = Av * rB;
      // alpha = a0*exp(-0.5*(C - A*tau)) = a0' * 2^(0.5*log2(e)*A*tau)
      float alpha = sc.x * fast_exp2(0.72134752f * (Av * tau));
      alpha = (tau > 0.1f) ? alpha : 0.0f;
      const float wgt = sT * alpha;
      sIr = fmaf(wgt, sc.y, sIr);
      sIg = fmaf(wgt, sc.z, sIg);
      sIb = fmaf(wgt, sc.w, sIb);
      sT  = fmaf(-alpha, sT, sT);                 // T *= (1 - alpha)
    }

    // merge half-segments in order: lo (g0-3) then hi (g4-7)
    const float oR = swapx16(sIr), oG = swapx16(sIg);
    const float oB = swapx16(sIb), oT = swapx16(sT);
    const float lR = hi ? oR : sIr, lG = hi ? oG : sIg;
    const float lB = hi ? oB : sIb, lT = hi ? oT : sT;
    const float hR = hi ? sIr : oR, hG = hi ? sIg : oG;
    const float hB = hi ? sIb : oB, hT = hi ? sT : oT;
    const float cR = fmaf(lT, hR, lR);
    const float cG = fmaf(lT, hG, lG);
    const float cB = fmaf(lT, hB, lB);
    const float cT = lT * hT;
    runR = fmaf(runT, cR, runR);
    runG = fmaf(runT, cG, runG);
    runB = fmaf(runT, cB, runB);
    runT *= cT;
  }

  if (!hi) {                                      // both halves hold identical state
    const int o = (y * GW + x) * 3;
    out[o + 0] = runR + runT;                     // reference adds T (white bg)
    out[o + 1] = runG + runT;
    out[o + 2] = runB + runT;
  }
}

extern "C" void kernel_launch(void* const* d_in, const int* in_sizes, int n_in,
                              void* d_out, int out_size, void* d_ws, size_t ws_size,
                              hipStream_t stream) {
  (void)in_sizes; (void)n_in; (void)out_size; (void)ws_size;
  const float* pos = (const float*)d_in[0];   // positions   [256,3]
  const float* lsc = (const float*)d_in[1];   // log_scales  [256,3]
  const float* rop = (const float*)d_in[2];   // raw_opacity [256]
  const float* col = (const float*)d_in[3];   // colors      [256,3]

  float4* aop  = (float4*)d_ws;               // 1024 float4 = 16 KB
  float4* scal = aop + NBLK * 32;             //  256 float4 =  4 KB

  gsplat_prep_kernel<<<1, 256, 0, stream>>>(pos, lsc, rop, col, aop, scal);

  dim3 grid(GW / 128, GH);                    // 8 waves/WG, 16 px/wave
  gsplat_render_kernel<<<grid, 256, 0, stream>>>(aop, scal, (float*)d_out);
}